// TwoDigitAdditionNetwork_78778290143909
// MI455X (gfx1250) — compile-verified
//
#include <hip/hip_runtime.h>

#define IN_SIZE   40
#define OUT_SIZE  22
#define HIDDEN    16384
#define FAN0      8192
#define FAN1      11
#define NTHREADS  512
#define HPT       (HIDDEN / NTHREADS)   // 32 hidden neurons per thread
#define PCOLS     24                    // 22 outputs padded to 24 (zeros)

typedef float v2f __attribute__((ext_vector_type(2)));
typedef float v8f __attribute__((ext_vector_type(8)));

// exp(-1/20) correctly rounded to f32 (matches jnp.exp(float32(-0.05)))
#define DECAY   0.95122942450071400910f
#define THRESH_ 0.3f

__global__ __launch_bounds__(NTHREADS, 1)
void snn_fused_kernel(const float* __restrict__ in_spk,
                      const float* __restrict__ w0,
                      const int*   __restrict__ tgt0,
                      const float* __restrict__ w1,
                      const int*   __restrict__ tgt1,
                      const int*   __restrict__ maxT,
                      float*       __restrict__ out)
{
    __shared__ float    sAcc[HIDDEN];          // add_h0, then nph0 (post-reset pot_h) : 64KB
    __shared__ float    sPart[NTHREADS][PCOLS];// per-thread output partials          : 48KB
    __shared__ unsigned sSpk[HIDDEN / 32];     // hidden spike bitmask                : 2KB
    __shared__ float    sOut[32];              // add_o1 (padded)
    __shared__ int      sAllFired;

    const int tid = threadIdx.x;
    const int T   = *maxT;

    // --- warm L2 for the phase-3 stream (overlaps with phases 1-2) -----------
    {
        const char* pw = (const char*)w1;
        const char* pt = (const char*)tgt1;
        const size_t bytes = (size_t)HIDDEN * FAN1 * sizeof(float);
        for (size_t off = (size_t)tid * 128; off < bytes; off += (size_t)NTHREADS * 128) {
            __builtin_prefetch(pw + off, 0, 1);
            __builtin_prefetch(pt + off, 0, 1);
        }
    }

    // --- init LDS -------------------------------------------------------------
    for (int k = 0; k < HPT; ++k) sAcc[tid + k * NTHREADS] = 0.0f;
    #pragma unroll
    for (int c = 0; c < PCOLS; ++c) sPart[tid][c] = 0.0f;
    if (tid < 32) sOut[tid] = 0.0f;
    __syncthreads();

    // --- phase 1: t=0 input scatter into hidden potentials (LDS atomics) -----
    // Active rows processed sequentially (barrier per row) -> fixed association.
    for (int i = 0; i < IN_SIZE; ++i) {
        const float s = in_spk[i];          // uniform across block
        if (s != 0.0f) {
            const float s2 = 2.0f * s;      // spk_in = input_spikes * 2
            const float* wrow = w0   + (size_t)i * FAN0;
            const int*   trow = tgt0 + (size_t)i * FAN0;
            for (int e = tid; e < FAN0; e += NTHREADS)
                atomicAdd(&sAcc[trow[e]], s2 * wrow[e]);
            __syncthreads();
        }
    }
    __syncthreads();

    // --- phase 2: hidden threshold at t=0; store post-reset pot and spike bits
    {
        unsigned mask = 0u;
        const int base = tid * 32;
        #pragma unroll 4
        for (int b = 0; b < 32; ++b) {
            const int h = base + b;
            const float p1 = sAcc[h] * DECAY;          // (pot + add)*decay at t=0
            const bool  sp = (p1 >= THRESH_);
            sAcc[h] = sp ? 0.0f : p1;                  // nph0 (reset on spike)
            mask |= sp ? (1u << b) : 0u;
        }
        sSpk[tid] = mask;
    }
    __syncthreads();

    // --- phase 3: hidden->output scatter into PRIVATE per-thread partial rows
    for (int k = 0; k < HPT; ++k) {
        const int h = tid + k * NTHREADS;
        if ((sSpk[h >> 5] >> (h & 31)) & 1u) {         // spk_h == 1 -> add w1 row
            const float* wr = w1   + (size_t)h * FAN1;
            const int*   tr = tgt1 + (size_t)h * FAN1;
            #pragma unroll
            for (int j = 0; j < FAN1; ++j)
                sPart[tid][tr[j]] += wr[j];            // no atomics: private row
        }
    }
    __syncthreads();

    // --- reduce 512 partial rows -> 64 rows (LDS tree, fixed order) ----------
    for (int s = NTHREADS / 2; s >= 64; s >>= 1) {
        for (int e = tid; e < s * PCOLS; e += NTHREADS) {
            const int r = e / PCOLS, c = e % PCOLS;
            sPart[r][c] += sPart[r + s][c];
        }
        __syncthreads();
    }

    // --- 64x22 column-sum via V_WMMA_F32_16X16X4_F32 on wave 0 ---------------
    // A = ones(16x4), B = 4x16 tile of partials, C accumulates column sums.
    if (tid < 32) {
        const int lane = tid;
        const int nn = lane & 15;       // N index
        const int kh = lane >> 4;       // lane-half -> K split
        v2f a; a[0] = 1.0f; a[1] = 1.0f;
        #pragma unroll
        for (int g = 0; g < 2; ++g) {
            const int col = g * 16 + nn;
            v8f acc = {};
            #pragma unroll
            for (int q = 0; q < 16; ++q) {             // 16 chunks of 4 rows
                const int r0 = q * 4 + 2 * kh;
                v2f b;
                b[0] = (col < PCOLS) ? sPart[r0 + 0][col] : 0.0f;
                b[1] = (col < PCOLS) ? sPart[r0 + 1][col] : 0.0f;
                acc = __builtin_amdgcn_wmma_f32_16x16x4_f32(
                          false, a, false, b, (short)0, acc, false, false);
            }
            // C row M=0 lives in acc[0] of lanes 0..15 (N = lane)
            if (kh == 0 && col < OUT_SIZE) sOut[col] = acc[0];
        }
    }
    __syncthreads();

    // --- closed-form finalization --------------------------------------------
    if (tid == 0) {
        int allf = 0;
        if (T >= 2) {
            allf = 1;
            for (int o = 0; o < OUT_SIZE; ++o) {
                const int f = (sOut[o] * DECAY >= THRESH_) ? 1 : 0; // fires at t=1
                allf &= f;
                out[o] = f ? 1.0f : -1.0f;                          // out_t
            }
            for (int o = 0; o < OUT_SIZE; ++o) {
                float p = sOut[o];
                const int n = allf ? 1 : (T - 1);   // frozen at t=1 vs decay to T-1
                for (int k = 0; k < n; ++k) p *= DECAY;   // sequential = bit-match
                out[OUT_SIZE + o] = p;                              // pot_o
            }
        } else {
            for (int o = 0; o < OUT_SIZE; ++o) { out[o] = -1.0f; out[OUT_SIZE + o] = 0.0f; }
        }
        sAllFired = allf;
    }
    __syncthreads();

    {
        const int allf = sAllFired;
        const int nd = (T >= 1) ? (allf ? 1 : (T - 1)) : 0;
        for (int k = 0; k < HPT; ++k) {
            const int h = tid + k * NTHREADS;
            float p = (T >= 1) ? sAcc[h] : 0.0f;
            for (int d = 0; d < nd; ++d) p *= DECAY;
            out[2 * OUT_SIZE + h] = p;                              // pot_h
        }
    }
}

extern "C" void kernel_launch(void* const* d_in, const int* in_sizes, int n_in,
                              void* d_out, int out_size, void* d_ws, size_t ws_size,
                              hipStream_t stream) {
    (void)in_sizes; (void)n_in; (void)out_size; (void)d_ws; (void)ws_size;
    const float* in_spk = (const float*)d_in[0];
    const float* w0     = (const float*)d_in[1];
    const int*   tgt0   = (const int*)  d_in[2];
    const float* w1     = (const float*)d_in[3];
    const int*   tgt1   = (const int*)  d_in[4];
    const int*   maxT   = (const int*)  d_in[5];
    float* out = (float*)d_out;

    snn_fused_kernel<<<dim3(1), dim3(NTHREADS), 0, stream>>>(
        in_spk, w0, tgt0, w1, tgt1, maxT, out);
}